// EdgeConv_687194767737
// MI455X (gfx1250) — compile-verified
//
#include <hip/hip_runtime.h>
#include <hip/hip_bf16.h>
#include <math.h>

typedef float v2f __attribute__((ext_vector_type(2)));
typedef float v8f __attribute__((ext_vector_type(8)));

#define NN   50000
#define KK   16
#define CC   128
#define OUTC 128

// ---------------------------------------------------------------------------
// Kernel A: Wd = W1 - W2   (W is 256x128 row-major; W1 = rows 0..127, W2 = rows 128..255)
// ---------------------------------------------------------------------------
__global__ __launch_bounds__(256) void wd_kernel(const float* __restrict__ W,
                                                 float* __restrict__ Wd) {
    int i = blockIdx.x * 256 + threadIdx.x;   // 16384 elements, 64 blocks
    Wd[i] = W[i] - W[i + CC * OUTC];
}

// ---------------------------------------------------------------------------
// Kernel B: fused dual GEMM via V_WMMA_F32_16X16X4_F32
//   y    = x @ W2          (N x 128)
//   base = x @ Wd + b      (N x 128)
// One block = 16 rows of x staged in LDS; 8 waves = 8 column tiles of 16.
// Each wave accumulates two 16x16 f32 tiles over 32 k-steps (K=4 each).
// ---------------------------------------------------------------------------
__global__ __launch_bounds__(256) void gemm_kernel(const float* __restrict__ x,
                                                   const float* __restrict__ W,
                                                   const float* __restrict__ Wd,
                                                   const float* __restrict__ b,
                                                   float* __restrict__ y,
                                                   float* __restrict__ base) {
    __shared__ float xs[16 * CC];           // 8 KB x-tile

    const int tid  = threadIdx.x;
    const int row0 = blockIdx.x * 16;

    // cooperative load of 16x128 f32 tile (512 float4, 256 threads -> 2 each)
    {
        const float4* src = (const float4*)(x + row0 * CC);
        float4* dst = (float4*)xs;
        dst[tid]       = src[tid];
        dst[tid + 256] = src[tid + 256];
    }
    __syncthreads();

    const int wave = tid >> 5;              // 0..7 -> output column tile
    const int lane = tid & 31;
    const int half = lane >> 4;             // 0 or 1
    const int lrow = lane & 15;
    const int koff = half * 2;              // lanes 0-15: K {0,1}; lanes 16-31: K {2,3}
    const int col0 = wave * 16;
    const int gc   = col0 + lrow;           // this lane's output column

    const float* __restrict__ W2 = W + CC * OUTC;   // rows 128..255 of W

    v8f acc_y = {0.f, 0.f, 0.f, 0.f, 0.f, 0.f, 0.f, 0.f};
    const float bv = b[gc];                 // bias depends only on column
    v8f acc_b = {bv, bv, bv, bv, bv, bv, bv, bv};

#pragma unroll 8
    for (int kk = 0; kk < 32; ++kk) {
        const int kb = kk * 4 + koff;
        // A fragment (16x4): this lane's row, two consecutive K values -> b64 LDS load
        v2f a = *(const v2f*)(&xs[lrow * CC + kb]);
        // B fragments (4x16): rows kb, kb+1 of W2 / Wd at column gc
        v2f b2; b2.x = W2[kb * OUTC + gc]; b2.y = W2[(kb + 1) * OUTC + gc];
        v2f bd; bd.x = Wd[kb * OUTC + gc]; bd.y = Wd[(kb + 1) * OUTC + gc];
        // (neg_a, A, neg_b, B, c_mod, C, reuse_a, reuse_b)
        acc_y = __builtin_amdgcn_wmma_f32_16x16x4_f32(false, a, false, b2,
                                                      (short)0, acc_y, false, false);
        acc_b = __builtin_amdgcn_wmma_f32_16x16x4_f32(false, a, false, bd,
                                                      (short)0, acc_b, false, false);
    }

    // C/D layout: VGPR r = row (half*8 + r), col = lane&15
    const int gr = row0 + half * 8;
#pragma unroll
    for (int r = 0; r < 8; ++r) {
        y[(gr + r) * OUTC + gc]    = acc_y[r];
        base[(gr + r) * OUTC + gc] = acc_b[r];
    }
}

// ---------------------------------------------------------------------------
// Kernel C: per-node gather-max + ELU.
//   out[i,:] = elu(base[i,:] + max over valid k of y[edge[i,k],:])
// One wave per node; each lane owns 4 channels (float4 -> 512B/row coalesced,
// served out of L2 since y (25.6MB) fits in the 192MB L2).
// ---------------------------------------------------------------------------
__global__ __launch_bounds__(256) void gather_max_kernel(const int* __restrict__ edge,
                                                         const float* __restrict__ y,
                                                         const float* __restrict__ base,
                                                         float* __restrict__ out) {
    const int wave = threadIdx.x >> 5;
    const int lane = threadIdx.x & 31;
    const int node = blockIdx.x * 8 + wave;
    if (node >= NN) return;

    const int c0 = lane * 4;
    const int* __restrict__ e = edge + node * KK;

    float4 m = make_float4(-INFINITY, -INFINITY, -INFINITY, -INFINITY);
#pragma unroll
    for (int k = 0; k < KK; ++k) {
        // edge id is wave-uniform: force it scalar so the skip branch is SALU
        const int j = __builtin_amdgcn_readfirstlane(e[k]);
        if (j >= 0) {
            const float4 v = *(const float4*)(y + j * OUTC + c0);
            m.x = fmaxf(m.x, v.x);
            m.y = fmaxf(m.y, v.y);
            m.z = fmaxf(m.z, v.z);
            m.w = fmaxf(m.w, v.w);
        }
    }

    const float4 bs = *(const float4*)(base + node * OUTC + c0);
    float4 z = make_float4(bs.x + m.x, bs.y + m.y, bs.z + m.z, bs.w + m.w);
    float4 o;
    o.x = (z.x > 0.f) ? z.x : (expf(z.x) - 1.f);
    o.y = (z.y > 0.f) ? z.y : (expf(z.y) - 1.f);
    o.z = (z.z > 0.f) ? z.z : (expf(z.z) - 1.f);
    o.w = (z.w > 0.f) ? z.w : (expf(z.w) - 1.f);
    *(float4*)(out + node * OUTC + c0) = o;
}

// ---------------------------------------------------------------------------
extern "C" void kernel_launch(void* const* d_in, const int* in_sizes, int n_in,
                              void* d_out, int out_size, void* d_ws, size_t ws_size,
                              hipStream_t stream) {
    const float* x    = (const float*)d_in[0];   // (N, C)
    const int*   edge = (const int*)  d_in[1];   // (N, K)
    const float* W    = (const float*)d_in[2];   // (2C, OUT)
    const float* b    = (const float*)d_in[3];   // (OUT,)
    float* out = (float*)d_out;

    float* ws   = (float*)d_ws;
    float* y    = ws;                        // N*128 floats = 25.6 MB
    float* base = ws + (size_t)NN * OUTC;    // N*128 floats = 25.6 MB
    float* Wd   = ws + (size_t)2 * NN * OUTC; // 128*128 floats = 64 KB

    // A: Wd = W1 - W2   (16384 elems)
    wd_kernel<<<64, 256, 0, stream>>>(W, Wd);
    // B: y = x@W2 ; base = x@Wd + b   (50000/16 = 3125 row tiles)
    gemm_kernel<<<NN / 16, 256, 0, stream>>>(x, W, Wd, b, y, base);
    // C: gather-max + elu   (8 nodes per block)
    gather_max_kernel<<<NN / 8, 256, 0, stream>>>(edge, y, base, out);
}